// PairwiseContactPredictionHead_4346506904008
// MI455X (gfx1250) — compile-verified
//
#include <hip/hip_runtime.h>

typedef __attribute__((ext_vector_type(2))) float v2f;
typedef __attribute__((ext_vector_type(8))) float v8f;

#define SEQ_L 384
#define HID   768
#define LOUT  382   // cropped output edge

// ---------------------------------------------------------------------------
// Prologue: Y0[j,h] = X[j,h] * W[h,0], Y1[j,h] = X[j,h] * W[h,1]
// (W is the (1536,2) matrix; Wp = W[:768] -> Wp[h,k] = W[2h + k])
// ---------------------------------------------------------------------------
__global__ __launch_bounds__(256) void scale_rows_kernel(
    const float* __restrict__ X, const float* __restrict__ W,
    float* __restrict__ Y0, float* __restrict__ Y1)
{
    int idx = blockIdx.x * blockDim.x + threadIdx.x;
    if (idx >= SEQ_L * HID) return;
    int h = idx % HID;
    float x = X[idx];
    Y0[idx] = x * W[2 * h + 0];
    Y1[idx] = x * W[2 * h + 1];
}

// ---------------------------------------------------------------------------
// Main GEMM: one wave32 per 16x16 output tile, C_k = X * Y_k^T via
// V_WMMA_F32_16X16X4_F32, K stepped by 4.
//
// 32-bit fragment layouts (ISA 7.12.2):
//   A 16x4 : lane l -> row M=l%16, K = 2*(l/16) + {VGPR0,VGPR1}
//   B 4x16 : lane l -> col N=l%16, K = 2*(l/16) + {VGPR0,VGPR1}  (B = Y^T)
//   C/D    : VGPR r -> M = r + 8*(l/16), N = l%16
// Both A and B fragments are contiguous float2 loads from row-major X / Y.
// ---------------------------------------------------------------------------
__global__ __launch_bounds__(32) void gram_wmma_kernel(
    const float* __restrict__ X,
    const float* __restrict__ Y0,
    const float* __restrict__ Y1,
    const float* __restrict__ bias,
    float* __restrict__ out)
{
    const int i0   = blockIdx.y * 16;
    const int j0   = blockIdx.x * 16;
    const int lane = threadIdx.x;     // 0..31
    const int hi   = lane >> 4;       // lane half: 0 or 1
    const int lm   = lane & 15;
    const int kb   = hi * 2;          // this lane's K sub-offset

    const float* arow  = X  + (i0 + lm) * HID + kb;
    const float* b0row = Y0 + (j0 + lm) * HID + kb;
    const float* b1row = Y1 + (j0 + lm) * HID + kb;

    v8f acc0 = {};
    v8f acc1 = {};

#pragma unroll 8
    for (int kk = 0; kk < HID; kk += 4) {
        v2f a  = *(const v2f*)(arow  + kk);
        v2f b0 = *(const v2f*)(b0row + kk);
        v2f b1 = *(const v2f*)(b1row + kk);
        acc0 = __builtin_amdgcn_wmma_f32_16x16x4_f32(
            false, a, false, b0, (short)0, acc0, false, false);
        acc1 = __builtin_amdgcn_wmma_f32_16x16x4_f32(
            false, a, false, b1, (short)0, acc1, false, false);
    }

    const float badd0 = bias[0];
    const float badd1 = bias[1];

#pragma unroll
    for (int r = 0; r < 8; ++r) {
        const int i = i0 + r + hi * 8;
        const int j = j0 + lm;
        if (i >= 1 && i <= LOUT && j >= 1 && j <= LOUT) {
            float2 v = make_float2(acc0[r] + badd0, acc1[r] + badd1);
            *(float2*)(out + ((size_t)(i - 1) * LOUT + (j - 1)) * 2) = v;
        }
    }
}

// ---------------------------------------------------------------------------
// Fallback GEMM (no workspace): fold W into the B fragment on the fly.
// W packed row-major (h,2): float4 at W + 2*(kk+kb) gives
// {Wp[k,0], Wp[k,1], Wp[k+1,0], Wp[k+1,1]} for k = kk+kb.
// ---------------------------------------------------------------------------
__global__ __launch_bounds__(32) void gram_wmma_fused_kernel(
    const float* __restrict__ X,
    const float* __restrict__ W,
    const float* __restrict__ bias,
    float* __restrict__ out)
{
    const int i0   = blockIdx.y * 16;
    const int j0   = blockIdx.x * 16;
    const int lane = threadIdx.x;
    const int hi   = lane >> 4;
    const int lm   = lane & 15;
    const int kb   = hi * 2;

    const float* arow = X + (i0 + lm) * HID + kb;
    const float* brow = X + (j0 + lm) * HID + kb;

    v8f acc0 = {};
    v8f acc1 = {};

#pragma unroll 8
    for (int kk = 0; kk < HID; kk += 4) {
        v2f a = *(const v2f*)(arow + kk);
        v2f x = *(const v2f*)(brow + kk);
        float4 w = *(const float4*)(W + 2 * (kk + kb));
        v2f b0 = { x.x * w.x, x.y * w.z };
        v2f b1 = { x.x * w.y, x.y * w.w };
        acc0 = __builtin_amdgcn_wmma_f32_16x16x4_f32(
            false, a, false, b0, (short)0, acc0, false, false);
        acc1 = __builtin_amdgcn_wmma_f32_16x16x4_f32(
            false, a, false, b1, (short)0, acc1, false, false);
    }

    const float badd0 = bias[0];
    const float badd1 = bias[1];

#pragma unroll
    for (int r = 0; r < 8; ++r) {
        const int i = i0 + r + hi * 8;
        const int j = j0 + lm;
        if (i >= 1 && i <= LOUT && j >= 1 && j <= LOUT) {
            float2 v = make_float2(acc0[r] + badd0, acc1[r] + badd1);
            *(float2*)(out + ((size_t)(i - 1) * LOUT + (j - 1)) * 2) = v;
        }
    }
}

extern "C" void kernel_launch(void* const* d_in, const int* in_sizes, int n_in,
                              void* d_out, int out_size, void* d_ws, size_t ws_size,
                              hipStream_t stream)
{
    (void)in_sizes; (void)n_in; (void)out_size;
    // setup_inputs order: inputs(f32), sequence_lengths(i32, unused), W(f32), b(f32)
    const float* X    = (const float*)d_in[0];
    const float* W    = (const float*)d_in[2];
    const float* bias = (const float*)d_in[3];
    float*       out  = (float*)d_out;

    const dim3 grid(SEQ_L / 16, SEQ_L / 16);  // 24 x 24 tiles
    const dim3 block(32);                     // one wave32 per tile

    const size_t ws_needed = (size_t)2 * SEQ_L * HID * sizeof(float);
    if (d_ws != nullptr && ws_size >= ws_needed) {
        float* Y0 = (float*)d_ws;
        float* Y1 = Y0 + (size_t)SEQ_L * HID;
        scale_rows_kernel<<<(SEQ_L * HID + 255) / 256, 256, 0, stream>>>(X, W, Y0, Y1);
        gram_wmma_kernel<<<grid, block, 0, stream>>>(X, Y0, Y1, bias, out);
    } else {
        gram_wmma_fused_kernel<<<grid, block, 0, stream>>>(X, W, bias, out);
    }
}